// SelectiveScanPeft_5136780886276
// MI455X (gfx1250) — compile-verified
//
#include <hip/hip_runtime.h>
#include <cstdint>
#include <cstddef>

// Problem dims (fixed by setup_inputs()).
#define BATCH  2
#define DDIM   1024
#define NSTATE 16
#define LSEQ   2048
#define BLOCK  256
#define TCH    (LSEQ / BLOCK)   // 8 timesteps per thread

__device__ __forceinline__ void load8(const float* __restrict__ p, float* r) {
    float4 v0 = reinterpret_cast<const float4*>(p)[0];
    float4 v1 = reinterpret_cast<const float4*>(p)[1];
    r[0] = v0.x; r[1] = v0.y; r[2] = v0.z; r[3] = v0.w;
    r[4] = v1.x; r[5] = v1.y; r[6] = v1.z; r[7] = v1.w;
}

__device__ __forceinline__ void store8(float* __restrict__ p, const float* r) {
    float4 v0 = make_float4(r[0], r[1], r[2], r[3]);
    float4 v1 = make_float4(r[4], r[5], r[6], r[7]);
    reinterpret_cast<float4*>(p)[0] = v0;
    reinterpret_cast<float4*>(p)[1] = v1;
}

__device__ __forceinline__ float softplus_f(float x) {
    // jax.nn.softplus = log1p(exp(x)); guard large x.
    return (x > 20.0f) ? x : __logf(1.0f + __expf(x));
}

__launch_bounds__(BLOCK, 1)
__global__ void selective_scan_kernel(const float* __restrict__ up,
                                      const float* __restrict__ dp,
                                      const float* __restrict__ Ap,
                                      const float* __restrict__ Bp,
                                      const float* __restrict__ Cp,
                                      const float* __restrict__ Dp,
                                      const float* __restrict__ zp,
                                      float* __restrict__ yp) {
    __shared__ float sA[NSTATE];
    __shared__ float sa[BLOCK][NSTATE + 1];   // +1 pad: kill LDS bank conflicts
    __shared__ float sh[BLOCK][NSTATE + 1];

    const int bd  = blockIdx.x;          // b*DDIM + d
    const int b   = bd / DDIM;
    const int d   = bd - b * DDIM;
    const int tid = threadIdx.x;
    const long rowBase = (long)bd * LSEQ;                 // u/delta/z/y
    const long bcBase  = (long)b * NSTATE * LSEQ;         // B/C
    const int  t0 = tid * TCH;

    // z is consumed last (phase 3): prefetch it now so it rides under the scan.
    __builtin_prefetch(zp + rowBase + t0, 0, 1);

    // Stage A[d,0:16] into LDS via the CDNA5 async-to-LDS path when available.
#if defined(__gfx1250__) && __has_builtin(__builtin_amdgcn_global_load_async_to_lds_b32)
    if (tid < NSTATE) {
        __builtin_amdgcn_global_load_async_to_lds_b32(
            (__attribute__((address_space(1))) int*)(Ap + (size_t)d * NSTATE + tid),
            (__attribute__((address_space(3))) int*)&sA[tid],
            0, 0);
    }
#if __has_builtin(__builtin_amdgcn_s_wait_asynccnt)
    __builtin_amdgcn_s_wait_asynccnt(0);
#else
    asm volatile("s_wait_asynccnt 0" ::: "memory");
#endif
#else
    if (tid < NSTATE) sA[tid] = Ap[(size_t)d * NSTATE + tid];
#endif
    const float Dd = Dp[d];
    __syncthreads();

    // ---------------- Phase 1: local chunk scan (the only exp pass) ----------
    float sp[TCH], uu[TCH], yloc[TCH];
    load8(dp + rowBase + t0, sp);
    load8(up + rowBase + t0, uu);
#pragma unroll
    for (int t = 0; t < TCH; ++t) {
        sp[t]   = softplus_f(sp[t]);
        yloc[t] = uu[t] * Dd;            // fold the skip connection in early
    }

    float G[NSTATE][TCH];                // C[n,t] * cumprod(a)[n,t]
    float aprod[NSTATE], hloc[NSTATE];   // per-chunk affine transform (A,H)

#pragma unroll
    for (int n = 0; n < NSTATE; ++n) {
        const float An = sA[n];
        float bb[TCH], cc[TCH];
        load8(Bp + bcBase + (long)n * LSEQ + t0, bb);
        load8(Cp + bcBase + (long)n * LSEQ + t0, cc);
        float P = 1.0f, h = 0.0f;
#pragma unroll
        for (int t = 0; t < TCH; ++t) {
            const float a = __expf(sp[t] * An);
            h = a * h + sp[t] * bb[t] * uu[t];   // local state (zero init)
            P *= a;
            yloc[t] += cc[t] * h;                // intra-chunk contribution
            G[n][t] = cc[t] * P;                 // prefix coupling coefficient
        }
        aprod[n] = P;
        hloc[n]  = h;
    }

    // ---------------- Phase 2: block scan of affine pairs (Hillis-Steele) ----
#pragma unroll
    for (int n = 0; n < NSTATE; ++n) { sa[tid][n] = aprod[n]; sh[tid][n] = hloc[n]; }
    __syncthreads();

    for (int off = 1; off < BLOCK; off <<= 1) {
        float pa[NSTATE], ph[NSTATE];
        const bool act = (tid >= off);
        if (act) {
#pragma unroll
            for (int n = 0; n < NSTATE; ++n) { pa[n] = sa[tid - off][n]; ph[n] = sh[tid - off][n]; }
        }
        __syncthreads();
        if (act) {
#pragma unroll
            for (int n = 0; n < NSTATE; ++n) {
                hloc[n]  = aprod[n] * ph[n] + hloc[n];  // self ∘ partner
                aprod[n] = aprod[n] * pa[n];
                sa[tid][n] = aprod[n];
                sh[tid][n] = hloc[n];
            }
        }
        __syncthreads();
    }

    float hpre[NSTATE];                  // state entering this thread's chunk
    if (tid == 0) {
#pragma unroll
        for (int n = 0; n < NSTATE; ++n) hpre[n] = 0.0f;
    } else {
#pragma unroll
        for (int n = 0; n < NSTATE; ++n) hpre[n] = sh[tid - 1][n];
    }

    // ---------------- Phase 3: prefix application + gate + store -------------
    float zz[TCH];
    load8(zp + rowBase + t0, zz);

    float y[TCH];
#pragma unroll
    for (int t = 0; t < TCH; ++t) {
        float acc = yloc[t];
#pragma unroll
        for (int n = 0; n < NSTATE; ++n) acc = __builtin_fmaf(G[n][t], hpre[n], acc);
        const float sig = 1.0f / (1.0f + __expf(-zz[t]));
        y[t] = acc * (zz[t] * sig);      // silu gate (skip already folded)
    }
    store8(yp + rowBase + t0, y);
}

extern "C" void kernel_launch(void* const* d_in, const int* in_sizes, int n_in,
                              void* d_out, int out_size, void* d_ws, size_t ws_size,
                              hipStream_t stream) {
    const float* u     = (const float*)d_in[0];
    const float* delta = (const float*)d_in[1];
    const float* A     = (const float*)d_in[2];
    const float* B     = (const float*)d_in[3];
    const float* C     = (const float*)d_in[4];
    const float* D     = (const float*)d_in[5];
    const float* z     = (const float*)d_in[6];
    float* y = (float*)d_out;

    dim3 grid(BATCH * DDIM);
    dim3 block(BLOCK);
    selective_scan_kernel<<<grid, block, 0, stream>>>(u, delta, A, B, C, D, z, y);
}